// DotApplier_82059645157826
// MI455X (gfx1250) — compile-verified
//
#include <hip/hip_runtime.h>
#include <cstdint>

#define IMG   1024
#define NDOTS 20
// LDS parameter layout (floats): [0,20) radius | [20,80) colors | [80,200) theta
#define NPARAM 200
#define DPW    10   // floats per precomputed active-dot record

__global__ __launch_bounds__(256)
void dot_applier_kernel(const float* __restrict__ radius,
                        const float* __restrict__ colors,
                        const float* __restrict__ theta,
                        float* __restrict__ out)
{
    __shared__ float sp[NPARAM];
    __shared__ float dpar[NDOTS * DPW];   // per active dot: t00,t10,hx,hy,rout2,invden,k0,k1,k2
    __shared__ int   dcount;
    const int tid = threadIdx.x;

    // ---- Stage the 200 uniform parameters into LDS via CDNA5 async-to-LDS ----
    if (tid < NPARAM) {
        const float* src;
        if (tid < 20)      src = radius + tid;
        else if (tid < 80) src = colors + (tid - 20);
        else               src = theta  + (tid - 80);
        // low 32 bits of a generic shared pointer == LDS byte offset
        uint32_t ldsOff = (uint32_t)(uintptr_t)(&sp[tid]);
        asm volatile("global_load_async_to_lds_b32 %0, %1, off"
                     :: "v"(ldsOff), "v"(src) : "memory");
    }
    asm volatile("s_wait_asynccnt 0" ::: "memory");
    __syncthreads();

    const int row  = blockIdx.x;
    const int col0 = tid << 2;

    const float invHalf = 2.0f / (float)IMG;        // base[i] = (i+0.5)*2/S - 1
    const float by = ((float)row + 0.5f) * invHalf - 1.0f;
    const float H = (float)IMG * 0.5f;              // 512
    const float CLMAX = (float)(IMG - 1);           // 1023

    // ---- parallel, order-preserving per-row dot compaction (wave 0) --------
    // Lane d tests whether dot d's warped footprint can touch this row
    // (gy range over bx in [-1,1] vs dot radius, in unclamped sample space;
    //  clamped border samples are always outside a centered dot of r<=~92px).
    // ballot + prefix-popcount keeps ascending dot order (last-dot-wins!).
    if (tid < 32) {
        bool flag = false;
        float t00 = 0.f, t10 = 0.f, hx = 0.f, hy = 0.f;
        float rout2 = 0.f, invden = 0.f, k0 = 0.f, k1 = 0.f, k2 = 0.f;
        if (tid < NDOTS) {
            const int d = tid;
            float rs   = sp[d] * (float)IMG;
            float rout = rs + 1.5f;               // bilinear corners <= sqrt(2) away
            t00 = sp[80 + 6*d + 0];
            float t01 = sp[80 + 6*d + 1];
            float t02 = sp[80 + 6*d + 2];
            t10 = sp[80 + 6*d + 3];
            float t11 = sp[80 + 6*d + 4];
            float t12 = sp[80 + 6*d + 5];
            hx = fmaf(t01, by, t02);              // row-invariant affine parts
            hy = fmaf(t11, by, t12);
            float a  = fabsf(t10);
            float lo = fmaf(hy - a, H, -0.5f) - H;   // min dcy over the row
            float hi = fmaf(hy + a, H, -0.5f) - H;   // max dcy over the row
            flag   = (hi >= -rout) && (lo <= rout);
            rout2  = rout * rout;
            invden = 1.0f / (rs * rs);            // hoisted divide (block-uniform)
            k0 = sp[20 + 3*d + 0] + 1.0f;
            k1 = sp[20 + 3*d + 1] + 1.0f;
            k2 = sp[20 + 3*d + 2] + 1.0f;
        }
#if __has_builtin(__builtin_amdgcn_ballot_w32)
        uint32_t mask = __builtin_amdgcn_ballot_w32(flag);
#else
        uint32_t mask = (uint32_t)__ballot(flag);
#endif
        if (flag) {
            int pos = __builtin_popcount(mask & ((1u << tid) - 1u));
            float* q = &dpar[pos * DPW];
            q[0] = t00; q[1] = t10; q[2] = hx; q[3] = hy;
            q[4] = rout2; q[5] = invden; q[6] = k0; q[7] = k1; q[8] = k2;
        }
        if (tid == 0) dcount = __builtin_popcount(mask);
    }
    __syncthreads();

    float bx[4];
#pragma unroll
    for (int p = 0; p < 4; ++p)
        bx[p] = ((float)(col0 + p) + 0.5f) * invHalf - 1.0f;

    float pr[4], pg[4], pb[4], pa[4];
#pragma unroll
    for (int p = 0; p < 4; ++p) { pr[p] = -1.0f; pg[p] = -1.0f; pb[p] = -1.0f; pa[p] = -1.0f; }

    const int nd = dcount;
    for (int j = 0; j < nd; ++j) {
        const float* q = &dpar[j * DPW];
        const float t00 = q[0], t10 = q[1], hx = q[2], hy = q[3];
        const float rout2 = q[4], invden = q[5], k0 = q[6], k1 = q[7], k2 = q[8];

        // ---- cheap prefix: unclamped center-relative coords + distance test ----
        // dcx = (gx+1)*512 - 0.5 - 512 = gx*512 - 0.5   (single FMA per axis)
        float dcx[4], dcy[4];
        bool anyIn = false;
#pragma unroll
        for (int p = 0; p < 4; ++p) {
            float gx = fmaf(t00, bx[p], hx);
            float gy = fmaf(t10, bx[p], hy);
            dcx[p] = fmaf(gx, H, -0.5f);
            dcy[p] = fmaf(gy, H, -0.5f);
            anyIn = anyIn || (fmaf(dcx[p], dcx[p], dcy[p] * dcy[p]) <= rout2);
        }
        // wave-uniform skip of dots this wave's 128-column span never touches
#if __has_builtin(__builtin_amdgcn_ballot_w32)
        if (__builtin_amdgcn_ballot_w32(anyIn) == 0u) continue;
#else
        if (!__any(anyIn)) continue;
#endif

        // ---- full bilinear + analytic dot evaluation ----
#pragma unroll
        for (int p = 0; p < 4; ++p) {
            float ix = fminf(fmaxf(dcx[p] + H, 0.0f), CLMAX);
            float iy = fminf(fmaxf(dcy[p] + H, 0.0f), CLMAX);
            float x0f = floorf(ix), y0f = floorf(iy);
            float wx = ix - x0f,    wy = iy - y0f;
            float x1f = fminf(x0f + 1.0f, CLMAX);
            float y1f = fminf(y0f + 1.0f, CLMAX);

            float dx0 = x0f - H, dx1 = x1f - H;
            float dy0 = y0f - H, dy1 = y1f - H;
            float sx0 = dx0 * dx0, sx1 = dx1 * dx1;
            float sy0 = dy0 * dy0, sy1 = dy1 * dy1;
            float cc[4] = { (sy0 + sx0) * invden, (sy0 + sx1) * invden,
                            (sy1 + sx0) * invden, (sy1 + sx1) * invden };

            float w[4];
            w[0] = (1.0f - wy) * (1.0f - wx);
            w[1] = (1.0f - wy) * wx;
            w[2] = wy * (1.0f - wx);
            w[3] = wy * wx;

            // per-corner: e = mask ? w : 0 ;  M = sum e ; Q = sum e*c^2
            // alpha: a = m*0.7*(1-0.8c^2) + (m-1)  =>  A = 1.7*M - 0.56*Q - 1
            float M = 0.0f, Q = 0.0f;
#pragma unroll
            for (int k = 0; k < 4; ++k) {
                float e  = (cc[k] <= 1.0f) ? w[k] : 0.0f;
                float c2 = cc[k] * cc[k];
                M += e;
                Q  = fmaf(e, c2, Q);
            }
            float A = fmaf(1.7f, M, fmaf(-0.56f, Q, -1.0f));

            // paint-if-nonnegative compositing (later dots override)
            if (A >= 0.0f) pa[p] = A;
            float r0 = fmaf(M, k0, -1.0f);   // = M*c - (1-M), since sum(w) = 1
            float r1 = fmaf(M, k1, -1.0f);
            float r2 = fmaf(M, k2, -1.0f);
            if (r0 >= 0.0f) pr[p] = r0;
            if (r1 >= 0.0f) pg[p] = r1;
            if (r2 >= 0.0f) pb[p] = r2;
        }
    }

    // ---- final clamp + vectorized b128 stores, one per output plane ----
    const int plane = IMG * IMG;
    const int base  = row * IMG + col0;
    float4 vR = make_float4(fmaxf(pr[0],0.f), fmaxf(pr[1],0.f), fmaxf(pr[2],0.f), fmaxf(pr[3],0.f));
    float4 vG = make_float4(fmaxf(pg[0],0.f), fmaxf(pg[1],0.f), fmaxf(pg[2],0.f), fmaxf(pg[3],0.f));
    float4 vB = make_float4(fmaxf(pb[0],0.f), fmaxf(pb[1],0.f), fmaxf(pb[2],0.f), fmaxf(pb[3],0.f));
    float4 vA = make_float4(fmaxf(pa[0],0.f), fmaxf(pa[1],0.f), fmaxf(pa[2],0.f), fmaxf(pa[3],0.f));
    *reinterpret_cast<float4*>(out + 0 * plane + base) = vR;
    *reinterpret_cast<float4*>(out + 1 * plane + base) = vG;
    *reinterpret_cast<float4*>(out + 2 * plane + base) = vB;
    *reinterpret_cast<float4*>(out + 3 * plane + base) = vA;
}

extern "C" void kernel_launch(void* const* d_in, const int* in_sizes, int n_in,
                              void* d_out, int out_size, void* d_ws, size_t ws_size,
                              hipStream_t stream) {
    // inputs: 0 adv_patch (unused), 1 alpha (unused), 2 radius, 3 colors, 4 theta
    const float* radius = (const float*)d_in[2];
    const float* colors = (const float*)d_in[3];
    const float* theta  = (const float*)d_in[4];
    float* out = (float*)d_out;   // [patch(3*S*S) | alpha(S*S)]

    dot_applier_kernel<<<dim3(IMG), dim3(256), 0, stream>>>(radius, colors, theta, out);
}